// TileWarping_87608742904185
// MI455X (gfx1250) — compile-verified
//
#include <hip/hip_runtime.h>
#include <hip/hip_bf16.h>
#include <stdint.h>

// Problem constants (fixed by the reference's setup_inputs)
#define UPF   4
#define BATCH 4
#define CCH   16
#define HH    512
#define WW    960
#define TH    (HH / UPF)   // 128
#define TW    (WW / UPF)   // 240

typedef uint32_t v4u_t __attribute__((ext_vector_type(4)));
typedef int      v8i_t __attribute__((ext_vector_type(8)));
typedef int      v4i_t __attribute__((ext_vector_type(4)));

// One workgroup handles one (b, y) full-resolution row.
// LDS holds fea_r[b, 0:16, y, 0:960]  (16*960*4 = 61440 bytes), staged by TDM.
__global__ __launch_bounds__(320) void tile_warp_kernel(
    const float* __restrict__ tp,   // tile_plane [B,3,TH,TW]
    const float* __restrict__ fl,   // fea_l      [B,C,H,W]
    const float* __restrict__ fr,   // fea_r      [B,C,H,W]
    float* __restrict__ out)        // output     [B,48,TH,TW]
{
    __shared__ float sR[CCH * WW];

    const int y   = blockIdx.x;   // 0..H-1
    const int b   = blockIdx.y;   // 0..B-1
    const int tid = threadIdx.x;  // 0..319

    // ---------------- Stage fea_r row (all 16 channels) into LDS ----------------
#if __has_builtin(__builtin_amdgcn_tensor_load_to_lds)
    if (tid < 32) {
        // Global address of tile start: fea_r[b, 0, y, 0]
        const uint64_t gaddr =
            (uint64_t)(const void*)(fr + ((size_t)b * CCH * HH + (size_t)y) * WW);
        const uint32_t lds = (uint32_t)(uint64_t)(void*)&sR[0];

        // D# group 0 (128b): count=1 | lds_addr | global_addr[56:0] | type=2
        v4u_t g0;
        g0.x = 1u;                                        // count=1, user-mode, no gather
        g0.y = lds;                                       // lds_addr (bytes)
        g0.z = (uint32_t)gaddr;                           // global_addr[31:0]
        g0.w = (uint32_t)(gaddr >> 32) | 0x80000000u;     // global_addr[56:32] | type=2<<30

        // D# group 1 (256b): 2D tile, data_size=4B, tile 960x16, stride H*W elems
        v8i_t g1;
        g1[0] = 0x20000;          // data_size=2 (4 bytes) at bits[17:16]; no pad/iter/mask
        g1[1] = (WW  << 16);      // tensor_dim0[15:0]  at bits[63:48]  = 960
        g1[2] = (CCH << 16);      // tensor_dim1[15:0]  at bits[95:80]  = 16
        g1[3] = (WW  << 16);      // tile_dim0          at bits[127:112] = 960
        g1[4] = CCH;              // tile_dim1          at bits[143:128] = 16 (tile_dim2=0)
        g1[5] = HH * WW;          // tensor_dim0_stride[31:0] (elements) = 491520
        g1[6] = 0;                // stride hi / tensor_dim1_stride lo
        g1[7] = 0;

        v4i_t gz = {0, 0, 0, 0};  // groups 2/3 unused (<=2D tensor)
#if defined(__clang_major__) && (__clang_major__ >= 23)
        v8i_t gz8 = {0, 0, 0, 0, 0, 0, 0, 0};
        __builtin_amdgcn_tensor_load_to_lds(g0, g1, gz, gz, gz8, 0);
#else
        __builtin_amdgcn_tensor_load_to_lds(g0, g1, gz, gz, 0);
#endif
        __builtin_amdgcn_s_wait_tensorcnt(0);             // s_wait_tensorcnt 0
    }
    __syncthreads();
#else
    // Fallback: cooperative coalesced staging
    for (int idx = tid; idx < CCH * WW; idx += 320) {
        const int c = idx / WW, x = idx - c * WW;
        sR[idx] = fr[(((size_t)b * CCH + c) * HH + (size_t)y) * WW + x];
    }
    __syncthreads();
#endif

    // ---------------- Compute 3 disparity hypotheses per pixel ----------------
    const int   i    = y & 3;        // sub-row inside tile
    const int   ty   = y >> 2;
    const float offi = (float)i - 1.5f;

    for (int x = tid; x < WW; x += 320) {
        const int   j    = x & 3;
        const int   tx   = x >> 2;
        const float offj = (float)j - 1.5f;

        // tile_plane: d, dx, dy for this tile
        const size_t tpi = (size_t)b * 3 * (TH * TW) + (size_t)ty * TW + tx;
        const float d  = tp[tpi];
        const float dx = tp[tpi + (size_t)(TH * TW)];
        const float dy = tp[tpi + (size_t)(2 * TH * TW)];

        // base disparity (disp_d = 0)
        const float disp = d + offi * dy + offj * dx;
        const float xf   = (float)x - disp;
        const float x0f  = floorf(xf);
        const float w1   = xf - x0f;          // shared fractional weight for all 3 hyps
        const float w0   = 1.0f - w1;
        const int   x0   = (int)x0f;

        // 4 consecutive taps x0-1 .. x0+2 cover all three hypotheses
        const int xm1 = x0 - 1, xp1 = x0 + 1, xp2 = x0 + 2;
        const float vm1 = (xm1 >= 0 && xm1 < WW) ? 1.0f : 0.0f;
        const float v0  = (x0  >= 0 && x0  < WW) ? 1.0f : 0.0f;
        const float v1  = (xp1 >= 0 && xp1 < WW) ? 1.0f : 0.0f;
        const float v2  = (xp2 >= 0 && xp2 < WW) ? 1.0f : 0.0f;
        const int cm1 = min(max(xm1, 0), WW - 1);
        const int c0i = min(max(x0,  0), WW - 1);
        const int c1i = min(max(xp1, 0), WW - 1);
        const int c2i = min(max(xp2, 0), WW - 1);

        float costA = 0.0f;  // disp_d = -1  -> taps (x0+1, x0+2)
        float costB = 0.0f;  // disp_d =  0  -> taps (x0,   x0+1)
        float costC = 0.0f;  // disp_d = +1  -> taps (x0-1, x0)

        const float* __restrict__ flp =
            fl + ((size_t)b * CCH * HH + (size_t)y) * WW + x;

#pragma unroll
        for (int c = 0; c < CCH; ++c) {
            const float fv = flp[(size_t)c * (HH * WW)];   // coalesced stream
            const float* row = &sR[c * WW];
            const float gm1 = row[cm1] * vm1;              // ds_load gathers
            const float g0v = row[c0i] * v0;
            const float g1v = row[c1i] * v1;
            const float g2v = row[c2i] * v2;
            costA += fabsf(fv - (g1v * w0 + g2v * w1));
            costB += fabsf(fv - (g0v * w0 + g1v * w1));
            costC += fabsf(fv - (gm1 * w0 + g0v * w1));
        }

        // out[b, blk*16 + i*4 + j, ty, tx]
        const int    ch = i * 4 + j;
        const size_t ob = (size_t)b * 48 * (TH * TW) + (size_t)ty * TW + tx;
        out[ob + (size_t)(ch)      * (TH * TW)] = costA;   // disp_d = -1
        out[ob + (size_t)(16 + ch) * (TH * TW)] = costB;   // disp_d =  0
        out[ob + (size_t)(32 + ch) * (TH * TW)] = costC;   // disp_d = +1
    }
}

extern "C" void kernel_launch(void* const* d_in, const int* in_sizes, int n_in,
                              void* d_out, int out_size, void* d_ws, size_t ws_size,
                              hipStream_t stream) {
    (void)in_sizes; (void)n_in; (void)out_size; (void)d_ws; (void)ws_size;
    const float* tp = (const float*)d_in[0];   // tile_plane [4,3,128,240]
    const float* fl = (const float*)d_in[1];   // fea_l      [4,16,512,960]
    const float* fr = (const float*)d_in[2];   // fea_r      [4,16,512,960]
    float* out = (float*)d_out;                // [4,48,128,240]

    dim3 grid(HH, BATCH);   // one block per (y, b) row
    dim3 block(320);        // 10 waves; 3 pixels per thread
    tile_warp_kernel<<<grid, block, 0, stream>>>(tp, fl, fr, out);
}